// HiPPOLayer_66657892434505
// MI455X (gfx1250) — compile-verified
//
#include <hip/hip_runtime.h>

// ---------------------------------------------------------------------------
// HiPPO layer for MI455X (gfx1250, wave32, WMMA).
//
// Roofline: out = 256 MB, x = 32 MB (read twice) + ~64 MB scratch traffic
//   => ~0.38 GB @ 23.3 TB/s ~ 16 us floor.  FLOPs ~ few GFLOP on the WMMA
//   pipe -> compute is free; the design goal is pure bandwidth + parallelism.
//
// Decomposition (L = 16):
//   Q_d = (A^T)^d,  r_d = b^T Q_d
//   s_{c+1} = s_c Q_16 + u_c,           u_c[n] = sum_j x_{16c+j} r_{15-j}[n]
//   c_{16c+j} = s_c Q_{j+1} + sum_{i<=j} x_{16c+i} r_{j-i}
// The second sum is a 16x16x16 lower-triangular-Toeplitz matmul ->
// four chained V_WMMA_F32_16X16X4_F32 per chunk.  The triangular mask is
// realized with a zero-padded LDS region so the A-matrix gather is
// branch-free; the boundary-term Q slice is chunk-invariant per lane and is
// hoisted into VGPRs once per wave.
// ---------------------------------------------------------------------------

typedef float v2f __attribute__((ext_vector_type(2)));
typedef float v8f __attribute__((ext_vector_type(8)));

#define NST    8
#define CHUNK  16
#define SEQ    4096
#define FEAT   64
#define BATCH  32
#define NCHUNK (SEQ / CHUNK)        // 256
#define BF     (BATCH * FEAT)       // 2048
#define NCHT   (BF * NCHUNK)        // 524288 chunks total

// workspace layout (float offsets)
#define WS_QPOW 0                          // Qpow[17][8][8]  (1088 floats)
#define WS_R    2048                       // R[16][8]        (128 floats)
#define WS_U    4096                       // U[NCHUNK][BF][8]
#define WS_SST  (4096 + (size_t)NCHT * NST)// S[NCHUNK][BF][8]
#define WS_FLOATS_NEEDED (4096 + 2 * (size_t)NCHT * NST)

// ---------------------------------------------------------------------------
// Kernel 0: precompute Q powers and r rows (one tiny block).
// ---------------------------------------------------------------------------
__global__ void hippo_precompute(const float* __restrict__ A,
                                 const float* __restrict__ Bv,
                                 float* __restrict__ ws) {
  __shared__ float q[64], qn[64], bsh[8];
  int t = threadIdx.x;            // 64 threads
  int m = t >> 3, n = t & 7;
  if (t < 8) bsh[t] = Bv[t];
  q[t] = (m == n) ? 1.0f : 0.0f;  // Q_0 = I
  __syncthreads();
  for (int d = 0; d <= 16; ++d) {
    ws[WS_QPOW + d * 64 + t] = q[t];
    if (d < 16 && m == 0) {       // r_d[n] = sum_m b[m] Q_d[m][n]
      float r = 0.0f;
      for (int k = 0; k < 8; ++k) r += bsh[k] * q[k * 8 + n];
      ws[WS_R + d * 8 + n] = r;
    }
    // Q_{d+1}[m][n] = sum_k Q_d[m][k] * (A^T)[k][n] = sum_k Q_d[m][k]*A[n][k]
    float acc = 0.0f;
    for (int k = 0; k < 8; ++k) acc += q[m * 8 + k] * A[n * 8 + k];
    qn[t] = acc;
    __syncthreads();
    q[t] = qn[t];
    __syncthreads();
  }
}

// ---------------------------------------------------------------------------
// Kernel 1 (Phase A): per-chunk input injection u_c. Fully parallel, coalesced.
// thread = (c, bf); x reads coalesced over f.
// ---------------------------------------------------------------------------
__global__ void __launch_bounds__(256)
hippo_chunk_u(const float* __restrict__ x, float* __restrict__ ws) {
  int tid = blockIdx.x * 256 + threadIdx.x;   // 0 .. NCHT-1
  int bf = tid & (BF - 1);
  int c  = tid >> 11;
  int b  = bf >> 6, f = bf & 63;
  const float* R = ws + WS_R;
  const float* xp = x + ((size_t)b * SEQ + (size_t)c * CHUNK) * FEAT + f;
  float u[8] = {0, 0, 0, 0, 0, 0, 0, 0};
#pragma unroll
  for (int j = 0; j < CHUNK; ++j) {
    float xv = xp[(size_t)j * FEAT];
    const float* r = R + (15 - j) * 8;        // uniform address -> s_load
#pragma unroll
    for (int n = 0; n < 8; ++n) u[n] = fmaf(xv, r[n], u[n]);
  }
  float* U = ws + WS_U + ((size_t)c * BF + bf) * 8;
  *(float4*)(U + 0) = make_float4(u[0], u[1], u[2], u[3]);
  *(float4*)(U + 4) = make_float4(u[4], u[5], u[6], u[7]);
}

// ---------------------------------------------------------------------------
// Kernel 2 (Phase B): chunk-boundary scan, 2048 lanes x 256 steps.
// U layout [c][bf][8] makes the per-step reads coalesced across lanes.
// ---------------------------------------------------------------------------
__global__ void __launch_bounds__(256)
hippo_scan(float* __restrict__ ws) {
  int bf = blockIdx.x * 256 + threadIdx.x;    // 0 .. 2047
  const float* Q16 = ws + WS_QPOW + 16 * 64;  // uniform -> scalar loads
  float q[64];
#pragma unroll
  for (int i = 0; i < 64; ++i) q[i] = Q16[i];
  float s[8] = {0, 0, 0, 0, 0, 0, 0, 0};
  const float* U = ws + WS_U;
  float* S = ws + WS_SST;
  for (int c = 0; c < NCHUNK; ++c) {
    size_t off = ((size_t)c * BF + bf) * 8;
    *(float4*)(S + off + 0) = make_float4(s[0], s[1], s[2], s[3]);
    *(float4*)(S + off + 4) = make_float4(s[4], s[5], s[6], s[7]);
    float4 u0 = *(const float4*)(U + off + 0);
    float4 u1 = *(const float4*)(U + off + 4);
    float u[8] = {u0.x, u0.y, u0.z, u0.w, u1.x, u1.y, u1.z, u1.w};
    float ns[8];
#pragma unroll
    for (int n = 0; n < 8; ++n) {
      float a = u[n];
#pragma unroll
      for (int m = 0; m < 8; ++m) a = fmaf(s[m], q[m * 8 + n], a);
      ns[n] = a;
    }
#pragma unroll
    for (int n = 0; n < 8; ++n) s[n] = ns[n];
  }
}

// ---------------------------------------------------------------------------
// Kernel 3 (Phase C): per-chunk outputs via WMMA Toeplitz matmul + boundary fix.
// Block = (b, c): stages x tile (with a leading zero pad of 16 rows) and
// Q_1..Q_16 in LDS; 8 waves each run 8 chunks (one per f).  Output is
// transposed through LDS -> fully coalesced float4 stores of the 256 MB out.
// ---------------------------------------------------------------------------
#define XROW 65                       // padded row stride (bank-conflict-free)
#define XPAD 16                       // leading zero rows for triangular mask

__global__ void __launch_bounds__(256)
hippo_out(const float* __restrict__ x, const float* __restrict__ ws,
          float* __restrict__ out) {
  int blk = blockIdx.x;                       // 0 .. BATCH*NCHUNK-1
  int b = blk >> 8;
  int c = blk & 255;

  __shared__ __align__(16) float xs[(XPAD + CHUNK) * XROW]; // [16 zero + 16 x]
  __shared__ __align__(16) float qs[16 * 64 + 64];          // Q_{j+1} (+pad)
  __shared__ __align__(16) float outs[CHUNK * FEAT * NST];  // 8192 [j][f*8+n]

  int t = threadIdx.x;

  // zero the pad region (rows 0..15); only rows 1..15 are ever read
  for (int i = t; i < XPAD * XROW; i += 256) xs[i] = 0.0f;

  // stage x tile (1024 contiguous floats, coalesced float4 loads)
  {
    const float* xbase = x + ((size_t)b * SEQ + (size_t)c * CHUNK) * FEAT;
    float4 v = ((const float4*)xbase)[t];
    int idx = t * 4;
    int j = idx >> 6, f0 = idx & 63;
    float* row = xs + (XPAD + j) * XROW + f0;
    row[0] = v.x; row[1] = v.y; row[2] = v.z; row[3] = v.w;
  }
  // stage Q_1..Q_16 (+ zero the tail pad so junk-lane reads are defined)
  for (int i = t; i < 1088; i += 256)
    qs[i] = (i < 1024) ? ws[WS_QPOW + 64 + i] : 0.0f;
  __syncthreads();

  int lane = t & 31;                // wave32
  int wave = t >> 5;
  int half = lane >> 4;             // lane half selects K pairs / M+8
  int jn   = lane & 15;             // A-row j (M) / B,D-column n (N)

  // B-matrix (r_d rows), wave-invariant: slice k, comp v -> K = 4k+2*half+v
  const float* R = ws + WS_R;
  v2f breg[4];
#pragma unroll
  for (int k = 0; k < 4; ++k) {
#pragma unroll
    for (int v = 0; v < 2; ++v) {
      int K = 4 * k + 2 * half + v;
      breg[k][v] = (jn < 8) ? R[K * 8 + jn] : 0.0f;   // cols 8..15 padded 0
    }
  }

  // Boundary-term Q slice is chunk-invariant per lane: hoist to VGPRs once.
  //   qreg[v*8+m] = Q_{j+1}[m][n],  j = v + 8*half, n = jn
  float qreg[64];
#pragma unroll
  for (int v = 0; v < 8; ++v) {
    int j = v + 8 * half;
#pragma unroll
    for (int m = 0; m < 8; ++m)
      qreg[v * 8 + m] = qs[j * 64 + m * 8 + jn];      // jn>=8: unused junk, stays in bounds
  }

  const float* Sbase = ws + WS_SST + ((size_t)c * BF + (size_t)b * 64) * 8;

  for (int i = 0; i < 8; ++i) {
    int f = wave * 8 + i;           // this wave's chunk (fixed b,c,f)

    // boundary state first: overlap global-load latency with the WMMA chain
    float s[8];
    {
      const float* sp = Sbase + (size_t)f * 8;
      float4 s0 = *(const float4*)(sp + 0);
      float4 s1 = *(const float4*)(sp + 4);
      s[0] = s0.x; s[1] = s0.y; s[2] = s0.z; s[3] = s0.w;
      s[4] = s1.x; s[5] = s1.y; s[6] = s1.z; s[7] = s1.w;
    }

    // A-matrix: lower-triangular Toeplitz of x, branch-free via zero pad:
    //   A[j][d] = xs[(j - d + XPAD)][f]   (rows < XPAD are zeros)
    v2f a[4];
#pragma unroll
    for (int k = 0; k < 4; ++k) {
#pragma unroll
      for (int v = 0; v < 2; ++v) {
        int d = 4 * k + 2 * half + v;
        a[k][v] = xs[(jn - d + XPAD) * XROW + f];
      }
    }

    // D = A(16x4) * B(4x16) + C, chained over K = 16
    v8f acc = {};
#pragma unroll
    for (int k = 0; k < 4; ++k) {
      acc = __builtin_amdgcn_wmma_f32_16x16x4_f32(
          /*neg_a=*/false, a[k], /*neg_b=*/false, breg[k],
          /*c_mod=*/(short)0, acc, /*reuse_a=*/false, /*reuse_b=*/false);
    }

    // boundary term: acc[v] += sum_m s_c[m] * Q_{j+1}[m][n] (registers only)
#pragma unroll
    for (int v = 0; v < 8; ++v) {
      float a2 = acc[v];
#pragma unroll
      for (int m = 0; m < 8; ++m)
        a2 = fmaf(s[m], qreg[v * 8 + m], a2);
      acc[v] = a2;
    }

    // park results in LDS for transposed coalesced store (only n<8 valid)
    if (jn < 8) {
#pragma unroll
      for (int v = 0; v < 8; ++v) {
        int j = v + 8 * half;
        outs[j * 512 + f * 8 + jn] = acc[v];
      }
    }
  }
  __syncthreads();

  // cooperative store: 16 rows x 512 floats, contiguous in out
  float4* dst = (float4*)(out + ((size_t)b * SEQ + (size_t)c * CHUNK) * 512);
  const float4* src = (const float4*)outs;
#pragma unroll
  for (int r = 0; r < 8; ++r) dst[r * 256 + t] = src[r * 256 + t];
}

// ---------------------------------------------------------------------------
// Fallback if workspace is too small: straightforward per-lane scan.
// ---------------------------------------------------------------------------
__global__ void __launch_bounds__(256)
hippo_naive(const float* __restrict__ x, const float* __restrict__ A,
            const float* __restrict__ Bv, float* __restrict__ out) {
  int bf = blockIdx.x * 256 + threadIdx.x;    // 0 .. 2047
  int b = bf >> 6, f = bf & 63;
  float At[64], bv[8];
#pragma unroll
  for (int i = 0; i < 64; ++i) At[i] = A[i];
#pragma unroll
  for (int i = 0; i < 8; ++i) bv[i] = Bv[i];
  float s[8] = {0, 0, 0, 0, 0, 0, 0, 0};
  const float* xp = x + (size_t)b * SEQ * FEAT + f;
  float* op = out + (size_t)b * SEQ * 512 + (size_t)f * 8;
  for (int tt = 0; tt < SEQ; ++tt) {
    float xv = xp[(size_t)tt * FEAT];
    float ns[8];
#pragma unroll
    for (int n = 0; n < 8; ++n) {
      float a = xv * bv[n];
#pragma unroll
      for (int m = 0; m < 8; ++m) a = fmaf(s[m], At[n * 8 + m], a);
      ns[n] = a;
    }
#pragma unroll
    for (int n = 0; n < 8; ++n) { s[n] = ns[n]; op[(size_t)tt * 512 + n] = ns[n]; }
  }
}

// ---------------------------------------------------------------------------
extern "C" void kernel_launch(void* const* d_in, const int* in_sizes, int n_in,
                              void* d_out, int out_size, void* d_ws, size_t ws_size,
                              hipStream_t stream) {
  const float* x  = (const float*)d_in[0];   // (32, 4096, 64) f32
  const float* Ad = (const float*)d_in[1];   // (8, 8) f32
  const float* Bd = (const float*)d_in[2];   // (8, 1) f32
  float* out = (float*)d_out;                // (32, 4096, 512) f32
  float* ws  = (float*)d_ws;

  if (ws_size < WS_FLOATS_NEEDED * sizeof(float)) {
    hippo_naive<<<BF / 256, 256, 0, stream>>>(x, Ad, Bd, out);
    return;
  }

  hippo_precompute<<<1, 64, 0, stream>>>(Ad, Bd, ws);
  hippo_chunk_u<<<NCHT / 256, 256, 0, stream>>>(x, ws);
  hippo_scan<<<BF / 256, 256, 0, stream>>>(ws);
  hippo_out<<<BATCH * NCHUNK, 256, 0, stream>>>(x, ws, out);
}